// ClassificationHead_33895881900609
// MI455X (gfx1250) — compile-verified
//
#include <hip/hip_runtime.h>

typedef float v2f __attribute__((ext_vector_type(2)));
typedef float v4f __attribute__((ext_vector_type(4)));
typedef float v8f __attribute__((ext_vector_type(8)));

#define TASKS 512
#define NWAY  5
#define NSHOT 5
#define NQ    75
#define NS    25
#define DIM   1600
#define PSTR  1604   // padded proto row stride (1604 % 64 == 4 -> avoids LDS bank conflicts)
#define ZROW  NWAY   // index of the all-zero prototype row (feeds B lanes n >= NWAY)

__device__ inline float dot4sq(v4f a) {
    return a.x * a.x + a.y * a.y + a.z * a.z + a.w * a.w;
}

__global__ __launch_bounds__(160)
void proto_head_kernel(const float* __restrict__ query,
                       const float* __restrict__ support,
                       const int*   __restrict__ slabels,
                       const float* __restrict__ scale,
                       float*       __restrict__ out)
{
    __shared__ float ldsProto[(NWAY + 1) * PSTR]; // 5 prototypes + 1 zero row (38,496 B)
    __shared__ float ldsAA[5 * 32];               // per-wave AA partials
    __shared__ float ldsBB[NWAY];                 // sum proto^2 per class
    __shared__ int   ldsLab[NS];

    const int task = blockIdx.x;
    const int tid  = threadIdx.x;
    const int lane = tid & 31;
    const int wave = tid >> 5;

    if (tid < NS)   ldsLab[tid] = slabels[task * NS + tid];
    if (tid < NWAY) ldsBB[tid]  = 0.0f;
    // zero row used by B lanes with n >= NWAY
    for (int c = tid; c < PSTR; c += 160) ldsProto[ZROW * PSTR + c] = 0.0f;
    __syncthreads();

    // ---------------- Phase 1: prototypes + BB ----------------
    const float* sup = support + (size_t)task * NS * DIM;
    float bb0 = 0.f, bb1 = 0.f, bb2 = 0.f, bb3 = 0.f, bb4 = 0.f;

    for (int c = tid; c < DIM / 4; c += 160) {
        v4f a0 = {0,0,0,0}, a1 = {0,0,0,0}, a2 = {0,0,0,0}, a3 = {0,0,0,0}, a4 = {0,0,0,0};
        for (int s = 0; s < NS; ++s) {
            v4f v = *reinterpret_cast<const v4f*>(sup + s * DIM + c * 4);
            switch (ldsLab[s]) {            // uniform per wave -> scalar branch
                case 0: a0 += v; break;
                case 1: a1 += v; break;
                case 2: a2 += v; break;
                case 3: a3 += v; break;
                default: a4 += v; break;
            }
        }
        const float inv = 1.0f / (float)NSHOT;
        a0.x *= inv; a0.y *= inv; a0.z *= inv; a0.w *= inv;
        a1.x *= inv; a1.y *= inv; a1.z *= inv; a1.w *= inv;
        a2.x *= inv; a2.y *= inv; a2.z *= inv; a2.w *= inv;
        a3.x *= inv; a3.y *= inv; a3.z *= inv; a3.w *= inv;
        a4.x *= inv; a4.y *= inv; a4.z *= inv; a4.w *= inv;
        *reinterpret_cast<v4f*>(&ldsProto[0 * PSTR + c * 4]) = a0;  bb0 += dot4sq(a0);
        *reinterpret_cast<v4f*>(&ldsProto[1 * PSTR + c * 4]) = a1;  bb1 += dot4sq(a1);
        *reinterpret_cast<v4f*>(&ldsProto[2 * PSTR + c * 4]) = a2;  bb2 += dot4sq(a2);
        *reinterpret_cast<v4f*>(&ldsProto[3 * PSTR + c * 4]) = a3;  bb3 += dot4sq(a3);
        *reinterpret_cast<v4f*>(&ldsProto[4 * PSTR + c * 4]) = a4;  bb4 += dot4sq(a4);
    }
    atomicAdd(&ldsBB[0], bb0);
    atomicAdd(&ldsBB[1], bb1);
    atomicAdd(&ldsBB[2], bb2);
    atomicAdd(&ldsBB[3], bb3);
    atomicAdd(&ldsBB[4], bb4);
    __syncthreads();

    // ---------------- Phase 2: WMMA AB + fused AA ----------------
    // Wave `wave` owns query rows [wave*16, wave*16+16).
    // A (16x4 f32): lanes 0-15 hold row=l16, K={k,k+1}; lanes 16-31 hold row=l16, K={k+2,k+3}.
    // B (4x16 f32): lanes 0-15 hold N=l16, K={k,k+1}; lanes 16-31 hold N=l16, K={k+2,k+3}.
    // Rows >= NQ are clamped to row 0 and loaded unconditionally (no EXEC divergence in
    // the hot loop); their D-rows / AA partials are simply never stored.
    const float sc   = scale[0];
    const int   tile = wave;
    const int   half = lane >> 4;       // 0 or 1 (selects K pair)
    const int   l16  = lane & 15;
    const int   arow = tile * 16 + l16;
    const int   crow = (arow < NQ) ? arow : 0;
    const float* qrow = query + (size_t)task * NQ * DIM + (size_t)crow * DIM + half * 2;
    const int   n     = l16;
    const int   bRow  = (n < NWAY) ? n : ZROW;   // lanes n>=NWAY read the zero row (broadcast)
    const float* bptr = &ldsProto[bRow * PSTR + half * 2];

    v8f acc0 = {0,0,0,0,0,0,0,0};
    v8f acc1 = {0,0,0,0,0,0,0,0};
    float aa = 0.0f;

    #pragma unroll 4
    for (int k = 0; k < DIM; k += 8) {
        v2f a0 = *reinterpret_cast<const v2f*>(qrow + k);
        v2f a1 = *reinterpret_cast<const v2f*>(qrow + k + 4);
        aa += a0.x * a0.x + a0.y * a0.y + a1.x * a1.x + a1.y * a1.y;

        v2f b0 = *reinterpret_cast<const v2f*>(bptr + k);
        v2f b1 = *reinterpret_cast<const v2f*>(bptr + k + 4);

        acc0 = __builtin_amdgcn_wmma_f32_16x16x4_f32(
                   false, a0, false, b0, (short)0, acc0, false, false);
        acc1 = __builtin_amdgcn_wmma_f32_16x16x4_f32(
                   false, a1, false, b1, (short)0, acc1, false, false);
    }

    ldsAA[wave * 32 + lane] = aa;
    __syncthreads();

    // ---------------- Epilogue ----------------
    // C/D layout: lanes 0-15: N=lane, VGPR j -> M=j; lanes 16-31: N=lane-16, VGPR j -> M=8+j.
    if (n < NWAY) {
        const float bbv = ldsBB[n];
        #pragma unroll
        for (int j = 0; j < 8; ++j) {
            const int m   = half * 8 + j;
            const int row = tile * 16 + m;
            if (row < NQ) {
                const float aarow = ldsAA[wave * 32 + m] + ldsAA[wave * 32 + m + 16];
                const float abv   = acc0[j] + acc1[j];
                const float logit = sc * (-(aarow - 2.0f * abv + bbv) * (1.0f / (float)DIM));
                out[((size_t)task * NQ + row) * NWAY + n] = logit;
            }
        }
    }
}

extern "C" void kernel_launch(void* const* d_in, const int* in_sizes, int n_in,
                              void* d_out, int out_size, void* d_ws, size_t ws_size,
                              hipStream_t stream) {
    const float* query   = (const float*)d_in[0];
    const float* support = (const float*)d_in[1];
    const int*   slab    = (const int*)  d_in[2];
    const float* scale   = (const float*)d_in[3];
    float*       out     = (float*)d_out;
    (void)in_sizes; (void)n_in; (void)out_size; (void)d_ws; (void)ws_size;

    proto_head_kernel<<<TASKS, 160, 0, stream>>>(query, support, slab, scale, out);
}